// S4Layer_12163347382742
// MI455X (gfx1250) — compile-verified
//
#include <hip/hip_runtime.h>

#define D_MODEL 512
#define D_STATE 64
#define BATCH   2
#define SEQLEN  4096
#define CH      32            // chunk length (uses A^32) -> 128 serial steps
#define NCHUNK  (SEQLEN / CH)
#define STR     65            // padded LDS stride for 64-wide matrices (bank-conflict free)
#define USTR    33            // padded stride for the 64x32 U matrix (odd -> conflict free)

typedef __attribute__((ext_vector_type(2))) float v2f;
typedef __attribute__((ext_vector_type(8))) float v8f;

// ---------------------------------------------------------------------------
// 64x64 fp32 matrix square in LDS via V_WMMA_F32_16X16X4_F32.
// One wave32, EXEC all ones. Dst = S * S. S/Dst padded with stride STR.
// Operand layouts per CDNA5 ISA 7.12.2:
//   A (16x4):  lanes 0-15 -> M=lane, VGPR0=K0,VGPR1=K1 ; lanes16-31 -> K2,K3
//   B (4x16):  lanes 0-15 -> N=lane, rows K0/K1 ; lanes16-31 -> rows K2/K3
//   C/D (16x16): VGPR v -> row v (lanes 0-15), row v+8 (lanes 16-31)
// ---------------------------------------------------------------------------
__device__ __forceinline__ void gemm64_sq_wmma(const float* S, float* Dst, int lane) {
  const int half = lane >> 4;      // 0 or 1
  const int l16  = lane & 15;
  for (int mt = 0; mt < 4; ++mt) {
    for (int nt = 0; nt < 4; ++nt) {
      v8f acc = {};
#pragma unroll
      for (int kc = 0; kc < 16; ++kc) {
        const int k0 = kc * 4 + 2 * half;
        const int am = mt * 16 + l16;
        const int bn = nt * 16 + l16;
        v2f a, b;
        a.x = S[am * STR + k0];
        a.y = S[am * STR + k0 + 1];
        b.x = S[k0 * STR + bn];
        b.y = S[(k0 + 1) * STR + bn];
        acc = __builtin_amdgcn_wmma_f32_16x16x4_f32(
            /*neg_a=*/false, a, /*neg_b=*/false, b,
            /*c_mod=*/(short)0, acc, /*reuse_a=*/false, /*reuse_b=*/false);
      }
#pragma unroll
      for (int v = 0; v < 8; ++v)
        Dst[(mt * 16 + v + 8 * half) * STR + nt * 16 + l16] = acc[v];
    }
  }
}

// ---------------------------------------------------------------------------
// Kernel 1: per-channel precompute (one wave32 per channel).
//   A32 = A^32 (5 WMMA squarings), W[m] = C A^{m+1}, U[:,31-m] = A^m B,
//   k[m] = C A^m B, for m = 0..31.
// ---------------------------------------------------------------------------
__global__ void __launch_bounds__(32)
s4_precompute(const float* __restrict__ A, const float* __restrict__ Bp,
              const float* __restrict__ Cp,
              float* __restrict__ wsA32, float* __restrict__ wsW,
              float* __restrict__ wsU, float* __restrict__ wsK) {
  __shared__ float b0[64 * STR];
  __shared__ float b1[64 * STR];
  __shared__ float Bs[64], Cs[64];
  __shared__ float vcur[64], vnxt[64];
  __shared__ float red[32];

  const int d = blockIdx.x;
  const int lane = threadIdx.x;

  for (int idx = lane; idx < 4096; idx += 32)
    b0[(idx >> 6) * STR + (idx & 63)] = A[d * 4096 + idx];
  Bs[lane]      = Bp[d * 64 + lane];
  Bs[lane + 32] = Bp[d * 64 + lane + 32];
  Cs[lane]      = Cp[d * 64 + lane];
  Cs[lane + 32] = Cp[d * 64 + lane + 32];
  __syncthreads();

  // ---- W rows (C A^{m+1}) and k_m = C A^m B, m = 0..31 ----
  vcur[lane] = Cs[lane]; vcur[lane + 32] = Cs[lane + 32];
  __syncthreads();
  for (int m = 0; m < CH; ++m) {
    red[lane] = vcur[lane] * Bs[lane] + vcur[lane + 32] * Bs[lane + 32];
    __syncthreads();
    if (lane == 0) {
      float s = 0.f;
      for (int i = 0; i < 32; ++i) s += red[i];
      wsK[d * CH + m] = s;
    }
    // vnxt = vcur @ A  (row-vector times matrix)
    for (int jj = 0; jj < 2; ++jj) {
      const int j = lane + jj * 32;
      float s = 0.f;
      for (int i = 0; i < 64; ++i) s += vcur[i] * b0[i * STR + j];
      vnxt[j] = s;
    }
    __syncthreads();
    vcur[lane] = vnxt[lane]; vcur[lane + 32] = vnxt[lane + 32];
    wsW[d * (CH * 64) + m * 64 + lane]      = vnxt[lane];
    wsW[d * (CH * 64) + m * 64 + lane + 32] = vnxt[lane + 32];
    __syncthreads();
  }

  // ---- U columns: U[:, 31-m] = A^m B ----
  vcur[lane] = Bs[lane]; vcur[lane + 32] = Bs[lane + 32];
  __syncthreads();
  for (int m = 0; m < CH; ++m) {
    if (m > 0) {
      for (int ii = 0; ii < 2; ++ii) {
        const int i = lane + ii * 32;
        float s = 0.f;
        for (int j = 0; j < 64; ++j) s += b0[i * STR + j] * vcur[j];
        vnxt[i] = s;
      }
      __syncthreads();
      vcur[lane] = vnxt[lane]; vcur[lane + 32] = vnxt[lane + 32];
      __syncthreads();
    }
    wsU[d * (64 * CH) + lane * CH + (CH - 1 - m)]        = vcur[lane];
    wsU[d * (64 * CH) + (lane + 32) * CH + (CH - 1 - m)] = vcur[lane + 32];
  }
  __syncthreads();

  // ---- A^32 by log-doubling with fp32 WMMA (5 squarings) ----
  gemm64_sq_wmma(b0, b1, lane); __syncthreads();  // A^2
  gemm64_sq_wmma(b1, b0, lane); __syncthreads();  // A^4
  gemm64_sq_wmma(b0, b1, lane); __syncthreads();  // A^8
  gemm64_sq_wmma(b1, b0, lane); __syncthreads();  // A^16
  gemm64_sq_wmma(b0, b1, lane); __syncthreads();  // A^32
  for (int idx = lane; idx < 4096; idx += 32)
    wsA32[d * 4096 + idx] = b1[(idx >> 6) * STR + (idx & 63)];
}

// ---------------------------------------------------------------------------
// Kernel 2: chunked scan. One workgroup per channel, 128 threads:
//   lanes  0-63  -> batch 0 state element i ; lanes 64-127 -> batch 1.
// Per chunk: h' = A32 h + U xi ; y_i = W_i.h + tri-conv(k, xi) + D xi_i.
// Ping-pong h buffers -> only 2 barriers per serial step.
// Prefetches next chunk's x column (global_prefetch_b8).
// Stores z = x + y for the LayerNorm pass.
// ---------------------------------------------------------------------------
__global__ void __launch_bounds__(128)
s4_scan(const float* __restrict__ x, const float* __restrict__ Dp,
        const float* __restrict__ wsA32, const float* __restrict__ wsW,
        const float* __restrict__ wsU, const float* __restrict__ wsK,
        float* __restrict__ z) {
  __shared__ float Am[64 * STR];
  __shared__ float Wm[CH * STR];
  __shared__ float Um[64 * USTR];
  __shared__ float kk[CH];
  __shared__ float h[2][2][64];   // [pingpong][batch][state]
  __shared__ float xc[2][CH];

  const int d = blockIdx.x;
  const int tid = threadIdx.x;
  const int b = tid >> 6;
  const int i = tid & 63;

  for (int idx = tid; idx < 4096; idx += 128)
    Am[(idx >> 6) * STR + (idx & 63)] = wsA32[d * 4096 + idx];
  for (int idx = tid; idx < CH * 64; idx += 128)
    Wm[(idx >> 6) * STR + (idx & 63)] = wsW[d * (CH * 64) + idx];
  for (int idx = tid; idx < 64 * CH; idx += 128)
    Um[(idx >> 5) * USTR + (idx & 31)] = wsU[d * (64 * CH) + idx];
  if (tid < CH) kk[tid] = wsK[d * CH + tid];
  h[0][b][i] = 0.f;
  const float Dd = Dp[d];
  __syncthreads();

  int p = 0;
  for (int c = 0; c < NCHUNK; ++c) {
    const int t0 = c * CH;
    if (tid < 64) {
      const int bb = tid >> 5, j = tid & 31;
      xc[bb][j] = x[((bb * SEQLEN) + t0 + j) * D_MODEL + d];
      if (c + 1 < NCHUNK)
        __builtin_prefetch(&x[((bb * SEQLEN) + t0 + CH + j) * D_MODEL + d], 0, 0);
    }
    __syncthreads();

    float acc = 0.f;
    for (int j = 0; j < 64; ++j) acc += Am[i * STR + j] * h[p][b][j];
    for (int j = 0; j < CH; ++j) acc += Um[i * USTR + j] * xc[b][j];
    h[p ^ 1][b][i] = acc;          // write opposite buffer: no WAR hazard

    if (i < CH) {
      float y = Dd * xc[b][i];
      for (int j = 0; j < 64; ++j) y += Wm[i * STR + j] * h[p][b][j];
      for (int j = 0; j <= i; ++j) y += kk[i - j] * xc[b][j];
      z[((b * SEQLEN) + t0 + i) * D_MODEL + d] = y + xc[b][i];  // residual
    }
    __syncthreads();
    p ^= 1;
  }
}

// ---------------------------------------------------------------------------
// Kernel 3: LayerNorm over D=512 per (b,t) row. 128 threads x float4.
// ---------------------------------------------------------------------------
__global__ void __launch_bounds__(128)
s4_layernorm(const float* __restrict__ z, const float* __restrict__ gamma,
             const float* __restrict__ beta, float* __restrict__ out) {
  __shared__ float rs[128], rq[128];
  const int row = blockIdx.x;
  const int tid = threadIdx.x;

  const float4 v = ((const float4*)(z + (size_t)row * D_MODEL))[tid];
  rs[tid] = v.x + v.y + v.z + v.w;
  rq[tid] = v.x * v.x + v.y * v.y + v.z * v.z + v.w * v.w;
  __syncthreads();
  for (int off = 64; off > 0; off >>= 1) {
    if (tid < off) { rs[tid] += rs[tid + off]; rq[tid] += rq[tid + off]; }
    __syncthreads();
  }
  const float mu  = rs[0] * (1.0f / D_MODEL);
  const float var = rq[0] * (1.0f / D_MODEL) - mu * mu;
  const float inv = rsqrtf(var + 1e-5f);

  const float4 g  = ((const float4*)gamma)[tid];
  const float4 be = ((const float4*)beta)[tid];
  float4 o;
  o.x = (v.x - mu) * inv * g.x + be.x;
  o.y = (v.y - mu) * inv * g.y + be.y;
  o.z = (v.z - mu) * inv * g.z + be.z;
  o.w = (v.w - mu) * inv * g.w + be.w;
  ((float4*)(out + (size_t)row * D_MODEL))[tid] = o;
}

// ---------------------------------------------------------------------------
extern "C" void kernel_launch(void* const* d_in, const int* in_sizes, int n_in,
                              void* d_out, int out_size, void* d_ws, size_t ws_size,
                              hipStream_t stream) {
  const float* x     = (const float*)d_in[0];
  const float* A     = (const float*)d_in[1];
  const float* Bp    = (const float*)d_in[2];
  const float* Cp    = (const float*)d_in[3];
  const float* Dp    = (const float*)d_in[4];
  const float* gamma = (const float*)d_in[5];
  const float* beta  = (const float*)d_in[6];

  float* ws    = (float*)d_ws;
  float* wsA32 = ws;                                   // 512*4096
  float* wsW   = wsA32 + (size_t)D_MODEL * 64 * 64;    // 512*32*64
  float* wsU   = wsW   + (size_t)D_MODEL * CH * 64;    // 512*64*32
  float* wsK   = wsU   + (size_t)D_MODEL * 64 * CH;    // 512*32
  float* wsZ   = wsK   + (size_t)D_MODEL * CH;         // 2*4096*512 (residual z)

  s4_precompute<<<D_MODEL, 32, 0, stream>>>(A, Bp, Cp, wsA32, wsW, wsU, wsK);
  s4_scan<<<D_MODEL, 128, 0, stream>>>(x, Dp, wsA32, wsW, wsU, wsK, wsZ);
  s4_layernorm<<<BATCH * SEQLEN, 128, 0, stream>>>(wsZ, gamma, beta, (float*)d_out);
}